// SMOTEmbed_36601711296977
// MI455X (gfx1250) — compile-verified
//
#include <hip/hip_runtime.h>

typedef float v2f __attribute__((ext_vector_type(2)));
typedef float v4f __attribute__((ext_vector_type(4)));
typedef float v8f __attribute__((ext_vector_type(8)));

constexpr int kT  = 8192;   // points
constexpr int kD  = 128;    // feature dim
constexpr int kNM = 4;      // synthetic multiplier
constexpr int kC  = 16;     // classes (output cols) -- exactly one WMMA N tile

// ---------------------------------------------------------------------------
// Kernel 1: row squared norms. 4 MB read, trivially bandwidth bound.
// ---------------------------------------------------------------------------
__global__ void __launch_bounds__(256) sq_kernel(const float* __restrict__ X,
                                                 float* __restrict__ sq) {
  int row = blockIdx.x * blockDim.x + threadIdx.x;
  if (row >= kT) return;
  const v4f* xp = (const v4f*)(X + (size_t)row * kD);
  float s = 0.0f;
#pragma unroll
  for (int t = 0; t < kD / 4; ++t) {
    v4f v = xp[t];
    s += v.x * v.x + v.y * v.y + v.z * v.z + v.w * v.w;
  }
  sq[row] = s;
}

// ---------------------------------------------------------------------------
// Kernel 2: fused Gram-matrix + top-2 neighbor search.
// One wave owns 16 "i" rows; their 16x128 tile lives in 64 VGPRs as the WMMA
// B operand for the whole kernel. For each 16-row "j" tile we accumulate
// S = Xj * Xi^T with 32 x v_wmma_f32_16x16x4_f32.  C/D layout: lane L holds
// column N = L%16 (an i row) for M = v + 8*(L/16) (j rows), so top-2 tracking
// is lane-private; halves merge with a width-32 shfl_xor at the end.
// Ranking key: sq_j - 2*S  (sq_i constant per row; sqrt monotonic).
// ---------------------------------------------------------------------------
__global__ void __launch_bounds__(256) knn_kernel(const float* __restrict__ X,
                                                  const float* __restrict__ sq,
                                                  int* __restrict__ nbr0,
                                                  int* __restrict__ nbr1) {
  const int lane = threadIdx.x & 31;
  const int wid  = blockIdx.x * (blockDim.x >> 5) + (threadIdx.x >> 5);
  const int i0   = wid * 16;
  const int lrow = lane & 15;
  const int hf   = lane >> 4;
  const int icol = i0 + lrow;

  // Pin B operand (this wave's 16 i-rows, all K) in registers: 32 x v2f.
  v2f bfrag[32];
  const float* bi = X + (size_t)(i0 + lrow) * kD + 2 * hf;
#pragma unroll
  for (int t = 0; t < 32; ++t) bfrag[t] = *(const v2f*)(bi + 4 * t);

  float b0d = __builtin_huge_valf(), b1d = __builtin_huge_valf();
  int   b0j = 0x7fffffff,            b1j = 0x7fffffff;

  for (int j0 = 0; j0 < kT; j0 += 16) {
    const float* aj = X + (size_t)(j0 + lrow) * kD + 2 * hf;
    if (j0 + 16 < kT) __builtin_prefetch(aj + 16 * kD, 0, 3);  // global_prefetch_b8

    v8f acc = {};
#pragma unroll
    for (int t = 0; t < 32; ++t) {
      v2f af = *(const v2f*)(aj + 4 * t);
      acc = __builtin_amdgcn_wmma_f32_16x16x4_f32(
          false, af, false, bfrag[t], (short)0, acc, false, false);
    }

#pragma unroll
    for (int v = 0; v < 8; ++v) {
      int j = j0 + v + 8 * hf;
      if (j == icol) continue;                 // exclude self
      float d = sq[j] - 2.0f * acc[v];
      if (d < b0d || (d == b0d && j < b0j)) {
        b1d = b0d; b1j = b0j; b0d = d; b0j = j;
      } else if (d < b1d || (d == b1d && j < b1j)) {
        b1d = d; b1j = j;
      }
    }
  }

  // Merge the two half-wave owners of each column (wave32: width 32).
  float o0d = __shfl_xor(b0d, 16, 32);
  int   o0j = __shfl_xor(b0j, 16, 32);
  float o1d = __shfl_xor(b1d, 16, 32);
  int   o1j = __shfl_xor(b1j, 16, 32);
  if (o0d < b0d || (o0d == b0d && o0j < b0j)) {
    b1d = b0d; b1j = b0j; b0d = o0d; b0j = o0j;
  } else if (o0d < b1d || (o0d == b1d && o0j < b1j)) {
    b1d = o0d; b1j = o0j;
  }
  if (o1d < b0d || (o1d == b0d && o1j < b0j)) {
    b1d = b0d; b1j = b0j; b0d = o1d; b0j = o1j;
  } else if (o1d < b1d || (o1d == b1d && o1j < b1j)) {
    b1d = o1d; b1j = o1j;
  }
  if (hf == 0) { nbr0[icol] = b0j; nbr1[icol] = b1j; }
}

// ---------------------------------------------------------------------------
// Kernel 3: logits = [X; syn] @ W + b, one 16-row tile per wave, N = kC = 16.
// Synthetic rows are materialized on the fly into the A fragment (never
// stored): a = x_src + g*(x_nbr - x_src).  W fragments preloaded to regs.
// ---------------------------------------------------------------------------
__global__ void __launch_bounds__(256) out_kernel(
    const float* __restrict__ X, const int* __restrict__ nn_idx,
    const float* __restrict__ gaps, const float* __restrict__ W,
    const float* __restrict__ bvec, const int* __restrict__ nbr0,
    const int* __restrict__ nbr1, float* __restrict__ out) {
  const int lane = threadIdx.x & 31;
  const int wid  = blockIdx.x * (blockDim.x >> 5) + (threadIdx.x >> 5);
  const int m0   = wid * 16;
  const int lrow = lane & 15;
  const int hf   = lane >> 4;

  // B operand = W (128x16): wfrag[t] = {W[k][n], W[k+1][n]}, k = 4t + 2*hf.
  v2f wf[32];
#pragma unroll
  for (int t = 0; t < 32; ++t) {
    int k   = 4 * t + 2 * hf;
    wf[t].x = W[(size_t)k * kC + lrow];
    wf[t].y = W[(size_t)(k + 1) * kC + lrow];
  }
  const float bias = bvec[lrow];

  const int row = m0 + lrow;
  v8f acc = {};
  if (m0 < kT) {                      // original rows (tiles never straddle kT)
    const float* ap = X + (size_t)row * kD + 2 * hf;
#pragma unroll
    for (int t = 0; t < 32; ++t) {
      v2f af = *(const v2f*)(ap + 4 * t);
      acc = __builtin_amdgcn_wmma_f32_16x16x4_f32(
          false, af, false, wf[t], (short)0, acc, false, false);
    }
  } else {                            // synthetic rows
    int   r = row - kT;
    int   s = r >> 2;                 // src = repeat(arange(T), 4)
    int   c = (nn_idx[r] == 0) ? nbr0[s] : nbr1[s];
    float g = gaps[r];
    const float* xs = X + (size_t)s * kD + 2 * hf;
    const float* xc = X + (size_t)c * kD + 2 * hf;
#pragma unroll
    for (int t = 0; t < 32; ++t) {
      v2f a0 = *(const v2f*)(xs + 4 * t);
      v2f a1 = *(const v2f*)(xc + 4 * t);
      v2f af;
      af.x = a0.x + g * (a1.x - a0.x);
      af.y = a0.y + g * (a1.y - a0.y);
      acc = __builtin_amdgcn_wmma_f32_16x16x4_f32(
          false, af, false, wf[t], (short)0, acc, false, false);
    }
  }

#pragma unroll
  for (int v = 0; v < 8; ++v) {
    int M = v + 8 * hf;
    out[(size_t)(m0 + M) * kC + lrow] = acc[v] + bias;
  }
}

// ---------------------------------------------------------------------------
extern "C" void kernel_launch(void* const* d_in, const int* in_sizes, int n_in,
                              void* d_out, int out_size, void* d_ws,
                              size_t ws_size, hipStream_t stream) {
  (void)in_sizes; (void)n_in; (void)out_size; (void)ws_size;
  const float* X      = (const float*)d_in[0];
  const int*   nn_idx = (const int*)d_in[1];
  const float* gaps   = (const float*)d_in[2];
  const float* W      = (const float*)d_in[3];
  const float* bvec   = (const float*)d_in[4];
  float*       out    = (float*)d_out;

  // Workspace: sq[kT] f32 | nbr0[kT] i32 | nbr1[kT] i32  (96 KB total)
  float* sq   = (float*)d_ws;
  int*   nbr0 = (int*)((char*)d_ws + (size_t)kT * sizeof(float));
  int*   nbr1 = nbr0 + kT;

  sq_kernel<<<kT / 256, 256, 0, stream>>>(X, sq);
  // 512 i-tiles, 8 waves (tiles) per 256-thread block -> 64 blocks
  knn_kernel<<<(kT / 16) / 8, 256, 0, stream>>>(X, sq, nbr0, nbr1);
  // (1 + kNM) * kT = 40960 rows -> 2560 tiles -> 320 blocks
  out_kernel<<<((1 + kNM) * kT / 16) / 8, 256, 0, stream>>>(
      X, nn_idx, gaps, W, bvec, nbr0, nbr1, out);
}